// GlobalConvolution_56487409877550
// MI455X (gfx1250) — compile-verified
//
#include <hip/hip_runtime.h>
#include <hip/hip_bf16.h>

// GlobalConvolution (FNO block): rfft(8192) -> per-head complex MLP -> irfft
// B=8, N=8192, C=512, H=8, DH=64, F=4097. fp32 WMMA 16x16x4 + TDM loads.

typedef __attribute__((ext_vector_type(2))) float v2f;
typedef __attribute__((ext_vector_type(8))) float v8f;
typedef __attribute__((ext_vector_type(4))) unsigned int u32x4;
typedef __attribute__((ext_vector_type(8))) int i32x8;
typedef __attribute__((ext_vector_type(4))) int i32x4;

__device__ __forceinline__ v8f wmma4(v2f a, v2f b, v8f c) {
  // D = A(16x4 f32) * B(4x16 f32) + C(16x16 f32)
  return __builtin_amdgcn_wmma_f32_16x16x4_f32(
      /*neg_a=*/false, a, /*neg_b=*/false, b,
      /*c_mod=*/(short)0, c, /*reuse_a=*/false, /*reuse_b=*/false);
}

__device__ __forceinline__ float gelu_exact(float v) {
  return 0.5f * v * (1.0f + erff(v * 0.70710678118f));
}

// ---------------------------------------------------------------------------
// Tensor Data Mover: 2D descriptor moving 4096 f32 elements between LDS
// (contiguous) and global memory (one element per 1024-element row, i.e. the
// even or odd samples of one channel of x[b][n][c]).  Issued by one wave.
// 6-arg builtin form (clang-23 / therock toolchain).
// ---------------------------------------------------------------------------
__device__ __forceinline__ void tdm_chan(bool store, unsigned lds_addr,
                                         unsigned long long gaddr) {
  u32x4 g0 = { 1u,                                   // count=1 (valid D#)
               lds_addr,                             // lds_addr (bytes)
               (unsigned)(gaddr & 0xffffffffull),    // global_addr[31:0]
               ((unsigned)((gaddr >> 32) & 0x01ffffffull)) | 0x80000000u }; // type=2
  i32x8 g1 = { 0x00020000,   // workgroup_mask=0, data_size=2 (4B)
               0x00010000,   // tensor_dim0 = 1
               0x10000000,   // tensor_dim1 = 4096
               0x00010000,   // tile_dim0 = 1
               0x00001000,   // tile_dim1 = 4096, tile_dim2 = 0
               1024,         // tensor_dim0_stride = 1024 elements (2 rows of C)
               0, 0 };       // tensor_dim1_stride = 0 (unused)
  i32x4 z4 = { 0, 0, 0, 0 };
  i32x8 z8 = { 0, 0, 0, 0, 0, 0, 0, 0 };
  if (store) __builtin_amdgcn_tensor_store_from_lds(g0, g1, z4, z4, z8, 0);
  else       __builtin_amdgcn_tensor_load_to_lds(g0, g1, z4, z4, z8, 0);
}

// ---------------------------------------------------------------------------
// Batched DFT-64 stage over a 64x64 complex matrix in LDS (in place).
// D[m][n] = sum_k A[m][k] * W[k][n] (complex); A read with strides (sm,sk),
// W from 64-entry cos/sin tables indexed (k*n)&63 (LDS broadcast, no
// conflicts).  Optional twiddle exp(sgn*2*pi*i*m*n/4096).  Result stored at
// mm + pout*n; pout=65 gives bank-conflict-free stores/next-stage reads.
// 8 waves x 2 tiles = 16 (16x16) tiles; per tile 16 K-steps x 4 WMMA.
// ---------------------------------------------------------------------------
__device__ __forceinline__ void dft64_stage(
    float* ZR, float* ZI, const float* CT, const float* ST,
    int sm, int sk, int pout, float sgn, bool twiddle,
    int wave, int l16, int hi) {
  float drs[2][8], dis[2][8];
  for (int t = 0; t < 2; ++t) {
    const int tile = wave + t * 8;
    const int Mb = (tile >> 2) << 4;
    const int Nb = (tile & 3) << 4;
    const int m = Mb + l16;
    const int n = Nb + l16;
    v8f prr = {0.f,0.f,0.f,0.f,0.f,0.f,0.f,0.f};
    v8f pii = {0.f,0.f,0.f,0.f,0.f,0.f,0.f,0.f};
    v8f pri = {0.f,0.f,0.f,0.f,0.f,0.f,0.f,0.f};
    v8f pir = {0.f,0.f,0.f,0.f,0.f,0.f,0.f,0.f};
    for (int k = 0; k < 64; k += 4) {
      const int k0 = k + (hi ? 2 : 0);
      v2f ar = { ZR[m * sm + k0 * sk],  ZR[m * sm + (k0 + 1) * sk] };
      v2f ai = { ZI[m * sm + k0 * sk],  ZI[m * sm + (k0 + 1) * sk] };
      v2f br = { CT[(k0 * n) & 63],     CT[((k0 + 1) * n) & 63] };
      v2f bi = { ST[(k0 * n) & 63],     ST[((k0 + 1) * n) & 63] };
      prr = wmma4(ar, br, prr);
      pii = wmma4(ai, bi, pii);
      pri = wmma4(ar, bi, pri);
      pir = wmma4(ai, br, pir);
    }
    for (int v = 0; v < 8; ++v) {
      float dr = prr[v] - pii[v];
      float di = pri[v] + pir[v];
      if (twiddle) {
        const int mm = Mb + v + (hi ? 8 : 0);
        const float ang = sgn * 1.5339808e-3f * (float)(mm * n);  // 2*pi/4096
        float s, c;
        __sincosf(ang, &s, &c);
        const float tr = dr * c - di * s;
        const float ti = dr * s + di * c;
        dr = tr; di = ti;
      }
      drs[t][v] = dr; dis[t][v] = di;
    }
  }
  __syncthreads();   // all reads of ZR/ZI complete
  for (int t = 0; t < 2; ++t) {
    const int tile = wave + t * 8;
    const int Mb = (tile >> 2) << 4;
    const int Nb = (tile & 3) << 4;
    const int n = Nb + l16;
    for (int v = 0; v < 8; ++v) {
      const int mm = Mb + v + (hi ? 8 : 0);
      ZR[mm + pout * n] = drs[t][v];
      ZI[mm + pout * n] = dis[t][v];
    }
  }
  __syncthreads();
}

// ---------------------------------------------------------------------------
// Kernel 1: forward rfft(8192, ortho) per (b,c) channel.  TDM gathers the
// strided channel (even/odd samples -> zr/zi) while the wave fills the DFT
// table; then 2 WMMA DFT-64 stages; untangle; write [b][h][f][dh] planes.
// ---------------------------------------------------------------------------
__global__ __launch_bounds__(256) void k_fft_fwd(
    const float* __restrict__ x, float* __restrict__ XR, float* __restrict__ XI) {
  __shared__ float zr[4160], zi[4160];   // 64x65 padded
  __shared__ float ct[64], st[64];
  const int b = blockIdx.x >> 9;
  const int c = blockIdx.x & 511;
  const int tid  = threadIdx.x;
  const int wave = tid >> 5;
  const int lane = tid & 31;
  const int l16  = lane & 15;
  const int hi   = lane >= 16;

  // async DMA: zr[k] = x[b][2k][c], zi[k] = x[b][2k+1][c]
  const float* xb = x + ((size_t)b * 8192u) * 512u + (size_t)c;
  if (tid < 32) {
    tdm_chan(false, (unsigned)(size_t)(void*)zr, (unsigned long long)(size_t)xb);
    tdm_chan(false, (unsigned)(size_t)(void*)zi, (unsigned long long)(size_t)(xb + 512));
  }
  // DFT-64 root table, overlapped with the DMA
  if (tid < 64) {
    const float ang = -9.8174770e-2f * (float)tid;  // -2*pi/64 * t
    ct[tid] = __cosf(ang);
    st[tid] = __sinf(ang);
  }
  if (tid < 32) __builtin_amdgcn_s_wait_tensorcnt(0);
  __syncthreads();

  // stage 1: A[m=n1][k=n2] = z[n1 + 64*n2]  -> twiddle -> pitch-65 store
  dft64_stage(zr, zi, ct, st, /*sm=*/1,  /*sk=*/64, /*pout=*/65, -1.f, true,  wave, l16, hi);
  // stage 2: A[m=k2][k=n1] at pitch 65     -> Z[k2+64*k1] at pitch 65
  dft64_stage(zr, zi, ct, st, /*sm=*/65, /*sk=*/1,  /*pout=*/65, -1.f, false, wave, l16, hi);

  // untangle rfft + ortho scale (0.5 from packing folded in)
  const int h = c >> 6, dh = c & 63;
  float* outR = XR + (((size_t)b * 8u + (size_t)h) * 4097u) * 64u + (size_t)dh;
  float* outI = XI + (((size_t)b * 8u + (size_t)h) * 4097u) * 64u + (size_t)dh;
  const float S = 0.5f * 0.011048543f;  // 0.5 / sqrt(8192)
  for (int f = tid; f <= 4096; f += 256) {
    const int j  = f & 4095;
    const int jm = (4096 - f) & 4095;
    const int ja  = j  + (j  >> 6);   // padded index
    const int jma = jm + (jm >> 6);
    const float ar = zr[ja],  ai = zi[ja];
    const float br = zr[jma], bi = -zi[jma];
    const float er = ar + br, ei = ai + bi;   // 2*E
    const float dr = ar - br, di = ai - bi;   // 2*D
    const float ang = -7.6699039e-4f * (float)f;  // -2*pi/8192 * f
    float s, cc;
    __sincosf(ang, &s, &cc);
    // X = E - i*W^f*D
    outR[(size_t)f * 64u] = S * (er + (cc * di + s * dr));
    outI[(size_t)f * 64u] = S * (ei - (cc * dr - s * di));
  }
}

// ---------------------------------------------------------------------------
// Kernel 2: per-head complex 2-layer MLP with exact GELU, in place on the
// frequency planes. One WG per (b, h, 64-frequency tile).
// ---------------------------------------------------------------------------
__global__ __launch_bounds__(256) void k_freq_mlp(
    float* __restrict__ XR, float* __restrict__ XI,
    const float* __restrict__ w1, const float* __restrict__ b1,
    const float* __restrict__ w2, const float* __restrict__ b2) {
  __shared__ float o1r[64 * 65], o1i[64 * 65];   // pitch-65: conflict-free
  const int F = 4097;
  int wg = blockIdx.x;
  const int ft = wg % 65; wg /= 65;
  const int h  = wg & 7;
  const int b  = wg >> 3;
  const int f0 = ft << 6;
  const int tid  = threadIdx.x;
  const int wave = tid >> 5;
  const int lane = tid & 31;
  const int l16  = lane & 15;
  const int hi   = lane >= 16;

  const size_t plane = (((size_t)b * 8u + (size_t)h) * (size_t)F) * 64u;
  float* xr = XR + plane;
  float* xi = XI + plane;
  const float* W1r = w1 + (size_t)h * 4096u;
  const float* W1i = w1 + (size_t)(8 + h) * 4096u;
  const float* W2r = w2 + (size_t)h * 4096u;
  const float* W2i = w2 + (size_t)(8 + h) * 4096u;
  const float* B1r = b1 + (size_t)h * 64u;
  const float* B1i = b1 + (size_t)(8 + h) * 64u;
  const float* B2r = b2 + (size_t)h * 64u;
  const float* B2i = b2 + (size_t)(8 + h) * 64u;

  // warm the (L2-resident) weight matrices: global_prefetch_b8
  if (tid < 128) {
    __builtin_prefetch(W1r + tid * 32, 0, 3);
    __builtin_prefetch(W1i + tid * 32, 0, 3);
    __builtin_prefetch(W2r + tid * 32, 0, 3);
    __builtin_prefetch(W2i + tid * 32, 0, 3);
  }

  // ---- layer 1: o1 = gelu(X * (W1r + i W1i) + b1), A streamed from global
  for (int t = 0; t < 2; ++t) {
    const int tile = wave + t * 8;
    const int Mb = (tile >> 2) << 4;
    const int Nb = (tile & 3) << 4;
    int fr = f0 + Mb + l16; if (fr >= F) fr = F - 1;   // row clamp (tail tile)
    const float* aR = xr + (size_t)fr * 64u;
    const float* aI = xi + (size_t)fr * 64u;
    const int n = Nb + l16;
    v8f prr = {0.f,0.f,0.f,0.f,0.f,0.f,0.f,0.f};
    v8f pii = {0.f,0.f,0.f,0.f,0.f,0.f,0.f,0.f};
    v8f pri = {0.f,0.f,0.f,0.f,0.f,0.f,0.f,0.f};
    v8f pir = {0.f,0.f,0.f,0.f,0.f,0.f,0.f,0.f};
    for (int k = 0; k < 64; k += 4) {
      const int k0 = k + (hi ? 2 : 0);
      v2f ar = { aR[k0], aR[k0 + 1] };
      v2f ai = { aI[k0], aI[k0 + 1] };
      v2f br = { W1r[k0 * 64 + n], W1r[(k0 + 1) * 64 + n] };
      v2f bi = { W1i[k0 * 64 + n], W1i[(k0 + 1) * 64 + n] };
      prr = wmma4(ar, br, prr);
      pii = wmma4(ai, bi, pii);
      pri = wmma4(ar, bi, pri);
      pir = wmma4(ai, br, pir);
    }
    const float br_ = B1r[n], bi_ = B1i[n];
    for (int v = 0; v < 8; ++v) {
      const int mm = Mb + v + (hi ? 8 : 0);
      o1r[mm * 65 + n] = gelu_exact(prr[v] - pii[v] + br_);
      o1i[mm * 65 + n] = gelu_exact(pri[v] + pir[v] + bi_);
    }
  }
  __syncthreads();

  // ---- layer 2: Y = o1 * (W2r + i W2i) + b2, written back in place
  for (int t = 0; t < 2; ++t) {
    const int tile = wave + t * 8;
    const int Mb = (tile >> 2) << 4;
    const int Nb = (tile & 3) << 4;
    const int n = Nb + l16;
    const int mrow = Mb + l16;
    v8f prr = {0.f,0.f,0.f,0.f,0.f,0.f,0.f,0.f};
    v8f pii = {0.f,0.f,0.f,0.f,0.f,0.f,0.f,0.f};
    v8f pri = {0.f,0.f,0.f,0.f,0.f,0.f,0.f,0.f};
    v8f pir = {0.f,0.f,0.f,0.f,0.f,0.f,0.f,0.f};
    for (int k = 0; k < 64; k += 4) {
      const int k0 = k + (hi ? 2 : 0);
      v2f ar = { o1r[mrow * 65 + k0], o1r[mrow * 65 + k0 + 1] };
      v2f ai = { o1i[mrow * 65 + k0], o1i[mrow * 65 + k0 + 1] };
      v2f br = { W2r[k0 * 64 + n], W2r[(k0 + 1) * 64 + n] };
      v2f bi = { W2i[k0 * 64 + n], W2i[(k0 + 1) * 64 + n] };
      prr = wmma4(ar, br, prr);
      pii = wmma4(ai, bi, pii);
      pri = wmma4(ar, bi, pri);
      pir = wmma4(ai, br, pir);
    }
    const float br_ = B2r[n], bi_ = B2i[n];
    for (int v = 0; v < 8; ++v) {
      const int mm = Mb + v + (hi ? 8 : 0);
      const int fg = f0 + mm;
      if (fg < F) {
        xr[(size_t)fg * 64u + n] = prr[v] - pii[v] + br_;
        xi[(size_t)fg * 64u + n] = pri[v] + pir[v] + bi_;
      }
    }
  }
}

// ---------------------------------------------------------------------------
// Kernel 3: irfft(8192, ortho) per (b,c) channel (mirror of kernel 1);
// TDM scatters the interleaved result back to y[b][n][c].
// ---------------------------------------------------------------------------
__global__ __launch_bounds__(256) void k_fft_inv(
    const float* __restrict__ YR, const float* __restrict__ YI,
    float* __restrict__ y) {
  __shared__ float zr[4160], zi[4160];
  __shared__ float ct[64], st[64];
  const int b = blockIdx.x >> 9;
  const int c = blockIdx.x & 511;
  const int tid  = threadIdx.x;
  const int wave = tid >> 5;
  const int lane = tid & 31;
  const int l16  = lane & 15;
  const int hi   = lane >= 16;
  const int h = c >> 6, dh = c & 63;

  const float* inR = YR + (((size_t)b * 8u + (size_t)h) * 4097u) * 64u + (size_t)dh;
  const float* inI = YI + (((size_t)b * 8u + (size_t)h) * 4097u) * 64u + (size_t)dh;

  // inverse untangle: Z[k] = E + i*W^k*D, W = exp(+2*pi*i/8192)
  for (int k = tid; k < 4096; k += 256) {
    const int km = 4096 - k;   // valid: 1..4096
    const float ar = inR[(size_t)k * 64u],  ai = inI[(size_t)k * 64u];
    const float br = inR[(size_t)km * 64u], bi = -inI[(size_t)km * 64u];
    const float er = ar + br, ei = ai + bi;
    const float dr = ar - br, di = ai - bi;
    const float ang = 7.6699039e-4f * (float)k;
    float s, cc;
    __sincosf(ang, &s, &cc);
    zr[k] = 0.5f * (er - (cc * di + s * dr));
    zi[k] = 0.5f * (ei + (cc * dr - s * di));
  }
  if (tid < 64) {
    const float ang = 9.8174770e-2f * (float)tid;  // +2*pi/64 * t (inverse)
    ct[tid] = __cosf(ang);
    st[tid] = __sinf(ang);
  }
  __syncthreads();

  dft64_stage(zr, zi, ct, st, /*sm=*/1,  /*sk=*/64, /*pout=*/65, +1.f, true,  wave, l16, hi);
  // final stage stores compact (pitch 64) so the TDM can stream it out
  dft64_stage(zr, zi, ct, st, /*sm=*/65, /*sk=*/1,  /*pout=*/64, +1.f, false, wave, l16, hi);

  // ortho scale in place: sqrt(8192)/4096
  const float S = 2.2097087e-2f;
  for (int k = tid; k < 4096; k += 256) {
    zr[k] *= S;
    zi[k] *= S;
  }
  __syncthreads();

  // async DMA scatter: y[b][2k][c] = zr[k], y[b][2k+1][c] = zi[k]
  float* yb = y + ((size_t)b * 8192u) * 512u + (size_t)c;
  if (tid < 32) {
    tdm_chan(true, (unsigned)(size_t)(void*)zr, (unsigned long long)(size_t)yb);
    tdm_chan(true, (unsigned)(size_t)(void*)zi, (unsigned long long)(size_t)(yb + 512));
    __builtin_amdgcn_s_wait_tensorcnt(0);
  }
}

// ---------------------------------------------------------------------------
extern "C" void kernel_launch(void* const* d_in, const int* in_sizes, int n_in,
                              void* d_out, int out_size, void* d_ws, size_t ws_size,
                              hipStream_t stream) {
  const float* x  = (const float*)d_in[0];
  const float* w1 = (const float*)d_in[1];
  const float* b1 = (const float*)d_in[2];
  const float* w2 = (const float*)d_in[3];
  const float* b2 = (const float*)d_in[4];
  float* y = (float*)d_out;

  const size_t plane = (size_t)8 * 8 * 4097 * 64;   // [B][H][F][DH] floats
  float* XR = (float*)d_ws;
  float* XI = XR + plane;

  k_fft_fwd <<<dim3(4096), dim3(256), 0, stream>>>(x, XR, XI);
  k_freq_mlp<<<dim3(4160), dim3(256), 0, stream>>>(XR, XI, w1, b1, w2, b2);
  k_fft_inv <<<dim3(4096), dim3(256), 0, stream>>>(XR, XI, y);
}